// Wav2Vec2GumbelVectorQuantizer_17763984736814
// MI455X (gfx1250) — compile-verified
//
#include <hip/hip_runtime.h>
#include <hip/hip_bf16.h>

typedef float v2f __attribute__((ext_vector_type(2)));
typedef float v8f __attribute__((ext_vector_type(8)));

#define N_TOK 24000   // B*S
#define HDIM  512
#define GV    640     // G*V
#define VPG   320     // V per group
#define DSUB  128     // D/G
#define ROWS  16      // token rows per block
#define NT    5       // 16-wide col tiles per wave (8 waves * 5 * 16 = 640)
#define APAD  20      // padded LDS stride for A chunk (conflict-free frag reads)

// ---------------------------------------------------------------------------
// Kernel 1: fused  logits-GEMM (fp32 WMMA)  +  gumbel-softmax-hard gather
//           +  per-block soft-softmax marginal partials
// ---------------------------------------------------------------------------
__global__ __launch_bounds__(256) void vq_fused_kernel(
    const float* __restrict__ hs,    // [N_TOK, 512]
    const float* __restrict__ gum,   // [N_TOK*2, 320]
    const float* __restrict__ W,     // [512, 640]
    const float* __restrict__ bias,  // [640]
    const float* __restrict__ cv,    // [640, 128]
    float*       __restrict__ out,   // [N_TOK, 256]
    float*       __restrict__ partial) // [1500, 640] per-block marginal sums
{
  __shared__ __align__(16) float lds_W[ROWS * GV];   // W k-chunk, reused for logits
  __shared__ __align__(16) float lds_A[ROWS * APAD]; // A k-chunk (16x16, padded)
  __shared__ float lds_marg[GV];

  const int tid  = threadIdx.x;
  const int lane = tid & 31;
  const int wave = tid >> 5;
  const int n0   = blockIdx.x * ROWS;

  v8f acc[NT] = {};

  const int arow = lane & 15;          // A row (M)
  const int kb   = (lane >> 4) << 1;   // K sub-offset per half-wave (0 or 2)
  const int colb = lane & 15;          // B col (N)

  for (int kc = 0; kc < HDIM; kc += 16) {
    __syncthreads();
    // Stage W[kc..kc+16) x 640 -> LDS (2560 float4, 10 per thread, coalesced)
    for (int i4 = tid; i4 < ROWS * (GV / 4); i4 += 256) {
      int r  = i4 / (GV / 4);
      int c4 = i4 - r * (GV / 4);
      float4 w = ((const float4*)(W + (size_t)(kc + r) * GV))[c4];
      ((float4*)lds_W)[r * (GV / 4) + c4] = w;
    }
    // Stage A[16 rows][kc..kc+16) -> LDS
    if (tid < 64) {
      int r = tid >> 2, c4 = tid & 3;
      float4 a = *(const float4*)(hs + (size_t)(n0 + r) * HDIM + kc + c4 * 4);
      *(float4*)&lds_A[r * APAD + c4 * 4] = a;
    }
    if (lane == 0 && kc + 16 < HDIM)   // pull next W chunk toward L2/WGP$
      __builtin_prefetch(W + (size_t)(kc + 16) * GV + wave * 320, 0, 1);
    __syncthreads();

    #pragma unroll
    for (int kk = 0; kk < 16; kk += 4) {
      v2f av;
      const float* ap = &lds_A[arow * APAD + kk + kb];
      av.x = ap[0];
      av.y = ap[1];
      #pragma unroll
      for (int t = 0; t < NT; ++t) {
        int col = (wave * NT + t) * 16 + colb;
        v2f bv;
        bv.x = lds_W[(kk + kb) * GV + col];
        bv.y = lds_W[(kk + kb + 1) * GV + col];
        acc[t] = __builtin_amdgcn_wmma_f32_16x16x4_f32(
            false, av, false, bv, (short)0, acc[t], false, false);
      }
    }
  }

  __syncthreads();
  // Write logits(+bias) tile into the now-dead W LDS region.
  {
    int rbase = (lane >> 4) * 8;   // C/D layout: lanes 16-31 hold M=8..15
    #pragma unroll
    for (int t = 0; t < NT; ++t) {
      int col = (wave * NT + t) * 16 + colb;
      float bv = bias[col];
      #pragma unroll
      for (int i = 0; i < 8; ++i)
        lds_W[(rbase + i) * GV + col] = acc[t][i] + bv;
    }
  }
  for (int c = tid; c < GV; c += 256) lds_marg[c] = 0.f;
  __syncthreads();

  // Phase 2: each wave handles 4 (row, group) pairs; 8 waves * 4 = 16 rows * 2 groups.
  #pragma unroll 1
  for (int q = 0; q < 4; ++q) {
    int pr  = wave * 4 + q;
    int row = pr >> 1;
    int g   = pr & 1;
    size_t n = (size_t)(n0 + row);
    const float* gp = gum + (n * 2 + g) * VPG;
    const float* lp = &lds_W[row * GV + g * VPG];

    float lloc[10], zloc[10];
    float lmax = -3.0e38f, zmax = -3.0e38f;
    int   zidx = 0x7fffffff;
    #pragma unroll
    for (int j = 0; j < 10; ++j) {
      int v   = lane + 32 * j;
      float l = lp[v];
      float z = (l + gp[v]) * 0.5f;   // /TAU, TAU=2
      lloc[j] = l;
      zloc[j] = z;
      lmax = fmaxf(lmax, l);
      if (z > zmax) { zmax = z; zidx = v; }  // keeps first max per lane
    }
    // wave32 reductions
    #pragma unroll
    for (int off = 16; off > 0; off >>= 1) {
      lmax = fmaxf(lmax, __shfl_xor(lmax, off, 32));
      float oz = __shfl_xor(zmax, off, 32);
      int   oi = __shfl_xor(zidx, off, 32);
      if (oz > zmax || (oz == zmax && oi < zidx)) { zmax = oz; zidx = oi; }
    }
    float lsum = 0.f, zsum = 0.f;
    #pragma unroll
    for (int j = 0; j < 10; ++j) {
      lsum += __expf(lloc[j] - lmax);
      zsum += __expf(zloc[j] - zmax);
    }
    #pragma unroll
    for (int off = 16; off > 0; off >>= 1) {
      lsum += __shfl_xor(lsum, off, 32);
      zsum += __shfl_xor(zsum, off, 32);
    }
    // non-gumbel softmax marginal accumulation (LDS atomics, block-local)
    float inv = 1.f / lsum;
    #pragma unroll
    for (int j = 0; j < 10; ++j) {
      int v = lane + 32 * j;
      atomicAdd(&lds_marg[g * VPG + v], __expf(lloc[j] - lmax) * inv);
    }
    // hard one-hot gather: probs[idx] = (1 - y_max) + y_max, rest exactly 0
    float ymax   = 1.f / zsum;
    float factor = (1.f - ymax) + ymax;
    float4 c4 = ((const float4*)(cv + (size_t)(g * VPG + zidx) * DSUB))[lane];
    float4 o4 = {factor * c4.x, factor * c4.y, factor * c4.z, factor * c4.w};
    ((float4*)(out + n * 256 + (size_t)g * DSUB))[lane] = o4;
  }

  __syncthreads();
  for (int c = tid; c < GV; c += 256)
    partial[(size_t)blockIdx.x * GV + c] = lds_marg[c];
}

// ---------------------------------------------------------------------------
// Kernel 2: reduce 1500 per-block marginal partials -> entropy term per column
// ---------------------------------------------------------------------------
__global__ __launch_bounds__(128) void vq_entropy_kernel(
    const float* __restrict__ partial, float* __restrict__ ent)
{
  __shared__ float red[128];
  int c = blockIdx.x;  // 0..639
  float s = 0.f;
  for (int b = threadIdx.x; b < 1500; b += 128)
    s += partial[(size_t)b * GV + c];
  red[threadIdx.x] = s;
  __syncthreads();
  for (int off = 64; off > 0; off >>= 1) {
    if (threadIdx.x < off) red[threadIdx.x] += red[threadIdx.x + off];
    __syncthreads();
  }
  if (threadIdx.x == 0) {
    float m = red[0] / 24000.f;
    ent[c] = -m * __logf(m + 1e-7f);
  }
}

// ---------------------------------------------------------------------------
// Kernel 3: perplexity scalar = sum_g exp(entropy_g)
// ---------------------------------------------------------------------------
__global__ void vq_perplexity_kernel(const float* __restrict__ ent,
                                     float* __restrict__ outp)
{
  if (threadIdx.x == 0) {
    float e0 = 0.f, e1 = 0.f;
    for (int i = 0; i < VPG; ++i) e0 += ent[i];
    for (int i = VPG; i < GV; ++i) e1 += ent[i];
    *outp = __expf(e0) + __expf(e1);
  }
}

extern "C" void kernel_launch(void* const* d_in, const int* in_sizes, int n_in,
                              void* d_out, int out_size, void* d_ws, size_t ws_size,
                              hipStream_t stream) {
  (void)in_sizes; (void)n_in; (void)out_size; (void)ws_size;
  const float* hs   = (const float*)d_in[0];
  const float* gum  = (const float*)d_in[1];
  const float* W    = (const float*)d_in[2];
  const float* bias = (const float*)d_in[3];
  const float* cv   = (const float*)d_in[4];
  float* out = (float*)d_out;

  float* partial = (float*)d_ws;                       // 1500*640 floats
  float* ent     = partial + (size_t)1500 * GV;        // 640 floats

  vq_fused_kernel<<<N_TOK / ROWS, 256, 0, stream>>>(hs, gum, W, bias, cv, out, partial);
  vq_entropy_kernel<<<GV, 128, 0, stream>>>(partial, ent);
  vq_perplexity_kernel<<<1, 32, 0, stream>>>(ent, out + (size_t)N_TOK * 256);
}